// UpDownMaxPooling_49452253446764
// MI455X (gfx1250) — compile-verified
//
#include <hip/hip_runtime.h>
#include <math.h>

// Cumulative max along H for x[B, 1, H, W], fp32. B*H*W elements in d_in[0],
// same count in d_out. Pure streaming scan: ~512 MiB traffic -> ~22us floor
// at 23.3 TB/s HBM. Strategy: one float4 column per thread, H tiled into
// 32-row tiles, double-buffered in LDS via gfx1250 async global->LDS loads
// (ASYNCcnt) so a full tile of loads is always in flight per wave.

typedef float v4f __attribute__((ext_vector_type(4)));

constexpr int kH = 1024;
constexpr int kW = 2048;
constexpr int kW4 = kW / 4;          // 512 float4 columns per row
constexpr int kColsPerBlock = 64;    // float4 columns per block == blockDim.x
constexpr int kTileRows = 32;        // rows per LDS tile
constexpr int kNumTiles = kH / kTileRows;

#if __has_builtin(__builtin_amdgcn_global_load_async_to_lds_b128) && \
    __has_builtin(__builtin_amdgcn_s_wait_asynccnt)
#define CUMMAX_ASYNC_PATH 1
#else
#define CUMMAX_ASYNC_PATH 0
#endif

#if CUMMAX_ASYNC_PATH
// Exact parameter types of the builtin (from clang's diagnostic):
//   param0: int __vector_size__(16) addrspace(1)*   (global source)
//   param1: int __vector_size__(16) addrspace(3)*   (LDS destination)
typedef int v4i_g __attribute__((vector_size(16)));
typedef __attribute__((address_space(1))) v4i_g* async_gptr_t;
typedef __attribute__((address_space(3))) v4i_g* async_lptr_t;
#endif

__device__ __forceinline__ v4f v4_max(v4f a, v4f b) {
  v4f r;
  r.x = fmaxf(a.x, b.x);
  r.y = fmaxf(a.y, b.y);
  r.z = fmaxf(a.z, b.z);
  r.w = fmaxf(a.w, b.w);
  return r;
}

__global__ __launch_bounds__(kColsPerBlock) void cummax_h_kernel(
    const float* __restrict__ in, float* __restrict__ out) {
#if CUMMAX_ASYNC_PATH
  // ---- Async double-buffered path (gfx1250 GLOBAL_LOAD_ASYNC_TO_LDS) ----
  __shared__ v4f tile[2][kTileRows * kColsPerBlock];

  const int t = threadIdx.x;
  const int blocksPerB = kW4 / kColsPerBlock;              // 8
  const int b = blockIdx.x / blocksPerB;
  const int c4 = (blockIdx.x % blocksPerB) * kColsPerBlock + t;

  const v4f* gin = (const v4f*)in + (size_t)b * kH * kW4 + c4;
  v4f* gout = (v4f*)out + (size_t)b * kH * kW4 + c4;

  // Issue all kTileRows async b128 loads for tile T into buffer `buf`.
  // Each lane's loads target the LDS bytes only that lane will read back,
  // so s_wait_asynccnt alone is a sufficient producer->consumer fence.
  auto issue_tile = [&](int T, int buf) {
    const v4f* src = gin + (size_t)T * kTileRows * kW4;
#pragma unroll
    for (int r = 0; r < kTileRows; ++r) {
      __builtin_amdgcn_global_load_async_to_lds_b128(
          (async_gptr_t)(src + (size_t)r * kW4),
          (async_lptr_t)&tile[buf][r * kColsPerBlock + t],
          /*offset=*/0, /*cpol=*/0);
    }
  };

  issue_tile(0, 0);

  v4f run = (v4f){-INFINITY, -INFINITY, -INFINITY, -INFINITY};

  for (int T = 0; T < kNumTiles; ++T) {
    const int buf = T & 1;
    if (T + 1 < kNumTiles) {
      issue_tile(T + 1, buf ^ 1);                  // keep next tile in flight
      __builtin_amdgcn_s_wait_asynccnt(kTileRows); // tile T done, T+1 pending
    } else {
      __builtin_amdgcn_s_wait_asynccnt(0);
    }

    v4f* dst = gout + (size_t)T * kTileRows * kW4;
#pragma unroll
    for (int r = 0; r < kTileRows; ++r) {
      v4f v = tile[buf][r * kColsPerBlock + t];
      run = v4_max(run, v);
      __builtin_nontemporal_store(run, dst + (size_t)r * kW4);
    }
  }
#else
  // ---- Fallback: register-resident scan, unrolled for MLP ----
  const int gid = blockIdx.x * kColsPerBlock + threadIdx.x;  // [0, B*kW4)
  const int b = gid / kW4;
  const int c4 = gid % kW4;

  const v4f* gin = (const v4f*)in + (size_t)b * kH * kW4 + c4;
  v4f* gout = (v4f*)out + (size_t)b * kH * kW4 + c4;

  v4f run = (v4f){-INFINITY, -INFINITY, -INFINITY, -INFINITY};
#pragma unroll 8
  for (int h = 0; h < kH; ++h) {
    if ((h & 7) == 0 && h + 16 < kH) {
      __builtin_prefetch(gin + (size_t)(h + 16) * kW4, 0, 0);
    }
    v4f v = gin[(size_t)h * kW4];
    run = v4_max(run, v);
    __builtin_nontemporal_store(run, gout + (size_t)h * kW4);
  }
#endif
}

extern "C" void kernel_launch(void* const* d_in, const int* in_sizes, int n_in,
                              void* d_out, int out_size, void* d_ws, size_t ws_size,
                              hipStream_t stream) {
  (void)n_in; (void)out_size; (void)d_ws; (void)ws_size;
  const float* in = (const float*)d_in[0];
  float* out = (float*)d_out;

  const int n = in_sizes[0];
  const int B = n / (kH * kW);                       // 32 for the reference
  const int blocksPerB = kW4 / kColsPerBlock;        // 8
  dim3 grid(B * blocksPerB);                         // 256 blocks
  dim3 block(kColsPerBlock);                         // 64 threads = 2 waves

  cummax_h_kernel<<<grid, block, 0, stream>>>(in, out);
}